// GraphSAGEModel_23364622090888
// MI455X (gfx1250) — compile-verified
//
#include <hip/hip_runtime.h>
#include <hip/hip_bf16.h>
#include <math.h>

typedef __attribute__((ext_vector_type(2))) float v2f;
typedef __attribute__((ext_vector_type(8))) float v8f;

#define N_NODES 50000
#define N_EDGES 800000
#define DIM     128
#define N_GRAPHS 64

// ---------------------------------------------------------------- fill zero
__global__ void gsage_fill0(float* __restrict__ p, int n) {
    int i = blockIdx.x * blockDim.x + threadIdx.x;
    int stride = gridDim.x * blockDim.x;
    for (; i < n; i += stride) p[i] = 0.0f;
}

// ------------------------------------------------- edge scatter (mean aggr)
// one wave32 per edge; lane handles 4 consecutive features (full 512B row).
__global__ void gsage_scatter(const float* __restrict__ x,
                              const int* __restrict__ src,
                              const int* __restrict__ dst,
                              float* __restrict__ agg,
                              float* __restrict__ deg,
                              int nEdges, int addDeg) {
    int g    = blockIdx.x * blockDim.x + threadIdx.x;
    int edge = g >> 5;
    int lane = g & 31;
    if (edge >= nEdges) return;
    int s = src[edge];
    int d = dst[edge];
    const float4 v = *(const float4*)(x + (size_t)s * DIM + lane * 4);
    float* p = agg + (size_t)d * DIM + lane * 4;
    __hip_atomic_fetch_add(p + 0, v.x, __ATOMIC_RELAXED, __HIP_MEMORY_SCOPE_AGENT);
    __hip_atomic_fetch_add(p + 1, v.y, __ATOMIC_RELAXED, __HIP_MEMORY_SCOPE_AGENT);
    __hip_atomic_fetch_add(p + 2, v.z, __ATOMIC_RELAXED, __HIP_MEMORY_SCOPE_AGENT);
    __hip_atomic_fetch_add(p + 3, v.w, __ATOMIC_RELAXED, __HIP_MEMORY_SCOPE_AGENT);
    if (addDeg && lane == 0)
        __hip_atomic_fetch_add(deg + d, 1.0f, __ATOMIC_RELAXED, __HIP_MEMORY_SCOPE_AGENT);
}

// --------------------------------------------------------- fused SAGE layer
// h[16 nodes x 128] = relu( (agg/deg) @ wl^T + bl + xin @ wr^T )
// block = 256 thr = 8 waves; wave w owns hidden cols [16w,16w+16)
// V_WMMA_F32_16X16X4_F32, K-loop over both weight matrices into one C frag.
#define LDSP 130  // row pad -> stride 2 banks, conflict-free 64b DS reads
__global__ __launch_bounds__(256)
void gsage_layer(const float* __restrict__ agg,
                 const float* __restrict__ deg,
                 const float* __restrict__ xin,
                 const float* __restrict__ wl,
                 const float* __restrict__ bl,
                 const float* __restrict__ wr,
                 float* __restrict__ hout) {
    __shared__ float sA[16][LDSP];   // degree-normalized aggregate
    __shared__ float sX[16][LDSP];   // root features
    __shared__ float sInv[16];

    const int tid      = threadIdx.x;
    const int nodeBase = blockIdx.x * 16;

    if (tid < 16) sInv[tid] = 1.0f / fmaxf(deg[nodeBase + tid], 1.0f);
    __syncthreads();

    for (int i = tid; i < 16 * DIM; i += 256) {
        int row = i >> 7, col = i & (DIM - 1);
        size_t off = (size_t)(nodeBase + row) * DIM + col;
        sA[row][col] = agg[off] * sInv[row];
        sX[row][col] = xin[off];
    }
    __syncthreads();

    const int wave  = tid >> 5;
    const int lane  = tid & 31;
    const int m     = lane & 15;     // node row inside tile / weight row
    const int hi    = lane >> 4;     // K sub-phase select
    const int nBase = wave * 16;

    const float* wlRow = wl + (size_t)(nBase + m) * DIM + 2 * hi;
    const float* wrRow = wr + (size_t)(nBase + m) * DIM + 2 * hi;

    v8f c = {};
#pragma unroll
    for (int k = 0; k < DIM; k += 4) {
        v2f a, b;
        a.x = sA[m][k + 2 * hi];
        a.y = sA[m][k + 2 * hi + 1];
        b.x = wlRow[k];
        b.y = wlRow[k + 1];
        c = __builtin_amdgcn_wmma_f32_16x16x4_f32(false, a, false, b,
                                                  (short)0, c, false, false);
    }
#pragma unroll
    for (int k = 0; k < DIM; k += 4) {
        v2f a, b;
        a.x = sX[m][k + 2 * hi];
        a.y = sX[m][k + 2 * hi + 1];
        b.x = wrRow[k];
        b.y = wrRow[k + 1];
        c = __builtin_amdgcn_wmma_f32_16x16x4_f32(false, a, false, b,
                                                  (short)0, c, false, false);
    }

    const float bias = bl[nBase + m];
#pragma unroll
    for (int r = 0; r < 8; ++r) {
        int row = r + 8 * hi;                          // D frag: M = r + 8*hi
        float v = c[r] + bias;
        hout[(size_t)(nodeBase + row) * DIM + nBase + m] = fmaxf(v, 0.0f);
    }
}

// ---------------------------------------------------------- global mean pool
__global__ void gsage_pool(const float* __restrict__ h,
                           const int* __restrict__ batch,
                           float* __restrict__ gsum,
                           float* __restrict__ gcnt) {
    int node = blockIdx.x;
    int c    = threadIdx.x;
    int b    = batch[node];
    __hip_atomic_fetch_add(gsum + (size_t)b * DIM + c,
                           h[(size_t)node * DIM + c],
                           __ATOMIC_RELAXED, __HIP_MEMORY_SCOPE_AGENT);
    if (c == 0)
        __hip_atomic_fetch_add(gcnt + b, 1.0f,
                               __ATOMIC_RELAXED, __HIP_MEMORY_SCOPE_AGENT);
}

// --------------------------------------------------------------- MLP head
__global__ void gsage_head(const float* __restrict__ gsum,
                           const float* __restrict__ gcnt,
                           const float* __restrict__ wl1,
                           const float* __restrict__ bl1,
                           const float* __restrict__ wout,
                           const float* __restrict__ bout,
                           float* __restrict__ out) {
    __shared__ float g[N_GRAPHS][DIM];
    int gi = threadIdx.x;                 // one thread per graph, 64 threads
    float invc = 1.0f / fmaxf(gcnt[gi], 1.0f);
    for (int k = 0; k < DIM; ++k)
        g[gi][k] = gsum[(size_t)gi * DIM + k] * invc;
    __syncthreads();

    float z = bout[0];
    for (int j = 0; j < 64; ++j) {
        float hj = bl1[j];
        const float* wrow = wl1 + (size_t)j * DIM;
        for (int k = 0; k < DIM; ++k) hj += g[gi][k] * wrow[k];
        hj = fmaxf(hj, 0.0f);
        z += hj * wout[j];
    }
    out[gi] = 1.0f / (1.0f + expf(-z));
}

// ---------------------------------------------------------------- launcher
extern "C" void kernel_launch(void* const* d_in, const int* in_sizes, int n_in,
                              void* d_out, int out_size, void* d_ws, size_t ws_size,
                              hipStream_t stream) {
    const float* x    = (const float*)d_in[0];
    const int*   ei   = (const int*)d_in[1];     // [2, N_EDGES] (jax x64 off -> int32)
    const int*   bat  = (const int*)d_in[2];
    const float* w1l  = (const float*)d_in[3];
    const float* b1l  = (const float*)d_in[4];
    const float* w1r  = (const float*)d_in[5];
    const float* w2l  = (const float*)d_in[6];
    const float* b2l  = (const float*)d_in[7];
    const float* w2r  = (const float*)d_in[8];
    const float* wl1  = (const float*)d_in[9];
    const float* bl1  = (const float*)d_in[10];
    const float* wout = (const float*)d_in[11];
    const float* bout = (const float*)d_in[12];
    (void)in_sizes; (void)n_in; (void)out_size; (void)ws_size;

    const int* src = ei;
    const int* dst = ei + N_EDGES;

    // workspace layout (floats): agg | h1 | deg | gsum | gcnt   (~51.4 MB)
    float* agg  = (float*)d_ws;
    float* h1   = agg + (size_t)N_NODES * DIM;
    float* deg  = h1 + (size_t)N_NODES * DIM;
    float* gsum = deg + N_NODES;
    float* gcnt = gsum + N_GRAPHS * DIM;
    float* h2   = agg;   // layer-2 output safely aliases agg (LDS-staged tiles)

    const int featN   = N_NODES * DIM;
    const int tailN   = N_NODES + N_GRAPHS * DIM + N_GRAPHS;  // deg|gsum|gcnt
    const int scatBlk = (N_EDGES * 32) / 256;                 // 100000
    const int gemmBlk = N_NODES / 16;                         // 3125 exact

    // zero accumulators
    gsage_fill0<<<2048, 256, 0, stream>>>(agg, featN);
    gsage_fill0<<<256, 256, 0, stream>>>(deg, tailN);

    // ---- layer 1
    gsage_scatter<<<scatBlk, 256, 0, stream>>>(x, src, dst, agg, deg, N_EDGES, 1);
    gsage_layer<<<gemmBlk, 256, 0, stream>>>(agg, deg, x, w1l, b1l, w1r, h1);

    // ---- layer 2
    gsage_fill0<<<2048, 256, 0, stream>>>(agg, featN);
    gsage_scatter<<<scatBlk, 256, 0, stream>>>(h1, src, dst, agg, deg, N_EDGES, 0);
    gsage_layer<<<gemmBlk, 256, 0, stream>>>(agg, deg, h1, w2l, b2l, w2r, h2);

    // ---- pool + head
    gsage_pool<<<N_NODES, DIM, 0, stream>>>(h2, bat, gsum, gcnt);
    gsage_head<<<1, N_GRAPHS, 0, stream>>>(gsum, gcnt, wl1, bl1, wout, bout,
                                           (float*)d_out);
}